// DGM_d_58308476011160
// MI455X (gfx1250) — compile-verified
//
#include <hip/hip_runtime.h>
#include <hip/hip_bf16.h>

#define N_NODES 4096
#define D_EMBED 64
#define F_IN    32
#define T_STEPS 24
#define KNN     8
#define PSTRIDE 4100   // padded LDS panel row stride (floats): 8-row offset = 32 banks

typedef __attribute__((ext_vector_type(2))) float v2f;
typedef __attribute__((ext_vector_type(8))) float v8f;

__device__ __forceinline__ float dev_temp(const float* tptr) {
    float t = tptr[0];
    t = fminf(5.0f, fmaxf(-5.0f, t));
    return expf(t);
}

// ---------------------------------------------------------------------------
// Kernel 1: temporal mean + embed GEMM. Writes x (both batches) to d_out,
// and for b==0 also x0 (row-major) and x0^T for WMMA B-fragments.
// 256 threads = 8 rows x 32 features.
// ---------------------------------------------------------------------------
__global__ void k_embed(const float* __restrict__ x_pre, const float* __restrict__ W,
                        float* __restrict__ out_x, float* __restrict__ x0,
                        float* __restrict__ x0T) {
    __shared__ float mf[8][F_IN];
    __shared__ float Ws[F_IN * D_EMBED];
    int t = threadIdx.x;
    for (int i = t; i < F_IN * D_EMBED; i += 256) Ws[i] = W[i];
    int lr = t >> 5;          // local row 0..7
    int f  = t & 31;          // feature 0..31
    long row = (long)blockIdx.x * 8 + lr;     // 0..8191
    int b = (int)(row >> 12);
    int n = (int)(row & 4095);
    float acc = 0.0f;
    const float* p = x_pre + (((long)b * T_STEPS) * N_NODES + n) * F_IN + f;
    for (int tt = 0; tt < T_STEPS; ++tt) acc += p[(long)tt * N_NODES * F_IN];
    mf[lr][f] = acc * (1.0f / T_STEPS);
    __syncthreads();
    float a0 = 0.0f, a1 = 0.0f;
    for (int ff = 0; ff < F_IN; ++ff) {
        float m = mf[lr][ff];
        a0 += m * Ws[ff * D_EMBED + f];
        a1 += m * Ws[ff * D_EMBED + f + 32];
    }
    long ob = ((long)b * N_NODES + n) * D_EMBED;
    out_x[ob + f]      = a0;
    out_x[ob + f + 32] = a1;
    if (b == 0) {
        x0[n * D_EMBED + f]      = a0;
        x0[n * D_EMBED + f + 32] = a1;
        x0T[f * N_NODES + n]        = a0;
        x0T[(f + 32) * N_NODES + n] = a1;
    }
}

// ---------------------------------------------------------------------------
// Kernel 1c: per-row squared norms of x0.
// ---------------------------------------------------------------------------
__global__ void k_sqnorm(const float* __restrict__ x0, float* __restrict__ sq) {
    int i = blockIdx.x * blockDim.x + threadIdx.x;
    if (i < N_NODES) {
        const float* r = x0 + i * D_EMBED;
        float s = 0.0f;
        for (int d = 0; d < D_EMBED; ++d) { float v = r[d]; s += v * v; }
        sq[i] = s;
    }
}

// ---------------------------------------------------------------------------
// Kernel 2: lq = max(0, s_i + s_j - 2 * x0 @ x0^T) * temp, via WMMA f32 16x16x4.
// 256 threads = 8 waves; each wave owns one 16x16 output tile (K = 64).
// ---------------------------------------------------------------------------
__global__ void k_lq(const float* __restrict__ x0, const float* __restrict__ x0T,
                     const float* __restrict__ sq, const float* __restrict__ tptr,
                     float* __restrict__ lq) {
    float temp = dev_temp(tptr);
    int wave = threadIdx.x >> 5;
    int lane = threadIdx.x & 31;
    int tile = blockIdx.x * 8 + wave;        // 0..65535
    int r0 = (tile >> 8) * 16;
    int c0 = (tile & 255) * 16;
    int m  = lane & 15;                      // A: row M / B,C: col N
    int hi = lane >> 4;
    int kb = hi * 2;                         // K sub-offset per lane group
    v8f c = {};
    for (int k0 = 0; k0 < D_EMBED; k0 += 4) {
        v2f a = *(const v2f*)(x0 + (r0 + m) * D_EMBED + k0 + kb);
        v2f bb;
        bb.x = x0T[(k0 + kb)     * N_NODES + c0 + m];
        bb.y = x0T[(k0 + kb + 1) * N_NODES + c0 + m];
        c = __builtin_amdgcn_wmma_f32_16x16x4_f32(false, a, false, bb,
                                                  (short)0, c, false, false);
    }
    float sj = sq[c0 + m];
#pragma unroll
    for (int v = 0; v < 8; ++v) {
        int mr = v + 8 * hi;                 // C layout: VGPR v -> M = v (+8 for hi lanes)
        float si = sq[r0 + mr];
        float d2 = fmaxf(si + sj - 2.0f * c[v], 0.0f) * temp;
        lq[(long)(r0 + mr) * N_NODES + c0 + m] = d2;
    }
}

// ---------------------------------------------------------------------------
// Kernel 2b: sqL[i] = sum_k lq[i][k]^2.
// ---------------------------------------------------------------------------
__global__ void k_rowsq(const float* __restrict__ lq, float* __restrict__ sqL) {
    __shared__ float red[256];
    int row = blockIdx.x;
    const float* r = lq + (long)row * N_NODES;
    float s = 0.0f;
    for (int j = threadIdx.x; j < N_NODES; j += 256) { float v = r[j]; s += v * v; }
    red[threadIdx.x] = s;
    __syncthreads();
    for (int off = 128; off > 0; off >>= 1) {
        if (threadIdx.x < off) red[threadIdx.x] += red[threadIdx.x + off];
        __syncthreads();
    }
    if (threadIdx.x == 0) sqL[row] = red[0];
}

// ---------------------------------------------------------------------------
// Kernel 3: fused G = lq @ lq (lq symmetric => row-major B loads),
// D2 = sL_i + sL_j - 2G streamed into a padded 16x4096 LDS panel,
// then two-level top-8 selection per row; emits edges + logprobs.
// Each wave processes 4 N-tiles per K-sweep: one A fragment feeds 4 WMMA
// (4 independent accumulator chains; ~6.4 FLOP/byte from L2).
// grid = 256 workgroups x 256 threads; one WG per WGP (~272KB dynamic LDS).
// ---------------------------------------------------------------------------
__global__ void k_gemm_topk(const float* __restrict__ lq, const float* __restrict__ sqL,
                            float* __restrict__ out) {
    extern __shared__ float smem[];
    float* d2p  = smem;                         // 16 * PSTRIDE floats
    float* cval = smem + 16 * PSTRIDE;          // 16 * 128 floats
    int*   cidx = (int*)(cval + 16 * 128);      // 16 * 128 ints

    int r0   = blockIdx.x * 16;
    int tid  = threadIdx.x;
    int wave = tid >> 5;
    int lane = tid & 31;
    int m  = lane & 15;
    int hi = lane >> 4;
    int kb = hi * 2;

    const float* arow = lq + (long)(r0 + m) * N_NODES;   // A rows for this lane

    // ---- GEMM phase: wave owns 32 contiguous N-tiles, 4 at a time.
    for (int p = 0; p < 8; ++p) {
        int n0 = wave * 512 + p * 64;           // 4 tiles: cols [n0, n0+64)
        v8f c0 = {}, c1 = {}, c2 = {}, c3 = {};
        for (int k0 = 0; k0 < N_NODES; k0 += 4) {
            v2f a = *(const v2f*)(arow + k0 + kb);
            const float* b0r = lq + (long)(k0 + kb)     * N_NODES + n0 + m;
            const float* b1r = lq + (long)(k0 + kb + 1) * N_NODES + n0 + m;
            v2f b;
            b.x = b0r[0];  b.y = b1r[0];
            c0 = __builtin_amdgcn_wmma_f32_16x16x4_f32(false, a, false, b,
                                                       (short)0, c0, false, false);
            b.x = b0r[16]; b.y = b1r[16];
            c1 = __builtin_amdgcn_wmma_f32_16x16x4_f32(false, a, false, b,
                                                       (short)0, c1, false, false);
            b.x = b0r[32]; b.y = b1r[32];
            c2 = __builtin_amdgcn_wmma_f32_16x16x4_f32(false, a, false, b,
                                                       (short)0, c2, false, false);
            b.x = b0r[48]; b.y = b1r[48];
            c3 = __builtin_amdgcn_wmma_f32_16x16x4_f32(false, a, false, b,
                                                       (short)0, c3, false, false);
        }
        // epilogue: D2 = sL_i + sL_j - 2G -> padded LDS panel
#pragma unroll
        for (int j = 0; j < 4; ++j) {
            int nj = n0 + j * 16 + m;
            float sj = sqL[nj];
            const v8f* cj = (j == 0) ? &c0 : (j == 1) ? &c1 : (j == 2) ? &c2 : &c3;
#pragma unroll
            for (int v = 0; v < 8; ++v) {
                int mr = v + 8 * hi;
                float si = sqL[r0 + mr];
                d2p[mr * PSTRIDE + nj] = fmaxf(si + sj - 2.0f * (*cj)[v], 0.0f);
            }
        }
    }
    __syncthreads();

    // ---- top-8 phase: 16 threads per row, strided local top-8 then merge
    int row = tid >> 4;      // 0..15
    int sub = tid & 15;
    float bv[8]; int bi[8];
#pragma unroll
    for (int j = 0; j < 8; ++j) { bv[j] = 3.4e38f; bi[j] = -1; }
    const float* dr = d2p + row * PSTRIDE;
    for (int j = sub; j < N_NODES; j += 16) {
        float v = dr[j];
        if (v < bv[7]) {
            int p = 7;
            while (p > 0 && bv[p - 1] > v) { bv[p] = bv[p - 1]; bi[p] = bi[p - 1]; --p; }
            bv[p] = v; bi[p] = j;
        }
    }
#pragma unroll
    for (int j = 0; j < 8; ++j) {
        cval[row * 128 + sub * 8 + j] = bv[j];
        cidx[row * 128 + sub * 8 + j] = bi[j];
    }
    __syncthreads();

    if (sub == 0) {
        float fv[8]; int fi[8];
#pragma unroll
        for (int j = 0; j < 8; ++j) { fv[j] = 3.4e38f; fi[j] = -1; }
        for (int cI = 0; cI < 128; ++cI) {
            float v = cval[row * 128 + cI];
            int  ix = cidx[row * 128 + cI];
            if (v < fv[7]) {
                int p = 7;
                while (p > 0 && fv[p - 1] > v) { fv[p] = fv[p - 1]; fi[p] = fi[p - 1]; --p; }
                fv[p] = v; fi[p] = ix;
            }
        }
        int gi = r0 + row;
        float* edge_src = out + (long)2 * N_NODES * D_EMBED;   // 524288
        float* edge_tgt = edge_src + N_NODES * KNN;            // +32768
        float* lp       = edge_tgt + N_NODES * KNN;            // +32768
        for (int j = 0; j < KNN; ++j) {
            int ix = fi[j];
            edge_src[gi * KNN + j] = (float)gi;
            edge_tgt[gi * KNN + j] = (float)ix;
            lp[gi * KNN + j]       = -lq[(long)gi * N_NODES + ix]; // temp already folded in
        }
    }
}

// ---------------------------------------------------------------------------
extern "C" void kernel_launch(void* const* d_in, const int* in_sizes, int n_in,
                              void* d_out, int out_size, void* d_ws, size_t ws_size,
                              hipStream_t stream) {
    const float* x_pre       = (const float*)d_in[0];
    const float* W_embed     = (const float*)d_in[1];
    const float* temperature = (const float*)d_in[2];
    float* out = (float*)d_out;

    // workspace layout (floats): x0 | x0T | sq | sqL | lq  (~69.3 MB total)
    float* ws  = (float*)d_ws;
    float* x0  = ws;
    float* x0T = x0  + N_NODES * D_EMBED;
    float* sq  = x0T + N_NODES * D_EMBED;
    float* sqL = sq  + N_NODES;
    float* lq  = sqL + N_NODES;

    k_embed <<<1024, 256, 0, stream>>>(x_pre, W_embed, out, x0, x0T);
    k_sqnorm<<<  16, 256, 0, stream>>>(x0, sq);
    k_lq    <<<8192, 256, 0, stream>>>(x0, x0T, sq, temperature, lq);
    k_rowsq <<<4096, 256, 0, stream>>>(lq, sqL);

    size_t shmem = (size_t)(16 * PSTRIDE + 16 * 128) * sizeof(float)
                 + (size_t)(16 * 128) * sizeof(int);   // ~272.6 KB < 320KB WGP LDS
    (void)hipFuncSetAttribute((const void*)k_gemm_topk,
                              hipFuncAttributeMaxDynamicSharedMemorySize, (int)shmem);
    k_gemm_topk<<<256, 256, shmem, stream>>>(lq, sqL, out);
}